// casConv2d_841813590169
// MI455X (gfx1250) — compile-verified
//
#include <hip/hip_runtime.h>

// ---------------------------------------------------------------------------
// casConv2d: 3x3 conv as blockwise-quantized GEMM, WMMA 16x16 tiles, wave32.
//   x: (4,32,32,32) f32 NCHW, w: (64,32,3,3) f32 -> out: (4,64,32,32) f32
//   K = 288 = 9 blocks of 32; accu per block, global 8-bit quant, sum blocks.
//
// Block = (b, oh): 8 waves cover 4 M-tiles x 2 N-tiles of one output row.
// LDS staging (dynamic shared, 124928 B):
//   wlds[64][292]: full weight matrix, float4 fill, stride 292 -> A-reads
//                  bank-conflict-free (32 lanes' b64 cover all 64 banks)
//   lds1[96][34] : halo'd 3-row input band (bounds checks paid once)
//   lds2[32][290]: im2col k-major per column, stride 290 -> same property
// Compute: per j-block, batch-load 8 A + 8 B fragments (ds_load_2addr_b64,
// imm offsets), then 8 back-to-back v_wmma_f32_16x16x4_f32 -> ~1 wait per j.
// Pass 1 -> global max|accu| (4B in d_ws, uint-bits atomicMax).
// Pass 2 -> recompute, quantize rintf(x/scale)*scale (RNE == jnp.round), sum.
// ---------------------------------------------------------------------------

typedef float v2f __attribute__((ext_vector_type(2)));
typedef float v8f __attribute__((ext_vector_type(8)));

#define C_IN  32
#define HW    32
#define KDIM  288                   // 32*3*3
#define OCH   64
#define LDIM  1024                  // 32*32
#define NBLK  9                     // real (non-zero-pad) 32-wide K blocks
#define LROW  34                    // halo'd row length (iw = -1..32)
#define NQ    (C_IN * 3)            // 96 rows (c*3+kh)
#define WSTR  292                   // weight row stride (pad: banks + f4 align)
#define BSTR  290                   // patch column stride (pad: banks)
#define WOFF  0                     // wlds: 64*292 floats
#define BOFF  (OCH * WSTR)          // lds2: 32*290 floats
#define HOFF  (BOFF + HW * BSTR)    // lds1: 96*34 floats
#define SMEM_FLOATS (HOFF + NQ * LROW)
#define SMEM_BYTES  (SMEM_FLOATS * 4)   // 124928

__device__ __forceinline__ void stage_all(float* wlds, float* lds1, float* lds2,
                                          const float* __restrict__ x,
                                          const float* __restrict__ w,
                                          int b, int oh) {
    // --- weights: 64 rows x 288, 4 threads/row, float4 copies, stride 292 ---
    {
        const int row = threadIdx.x >> 2;          // 0..63
        const int q   = (threadIdx.x & 3) * 72;    // quarter-row offset
        const float* src = w + row * KDIM + q;
        float* dst = wlds + row * WSTR + q;
#pragma unroll
        for (int i = 0; i < 18; ++i)
            *(float4*)(dst + i * 4) = *(const float4*)(src + i * 4);
    }
    // --- halo'd input band: lds1[(c*3+kh)*34 + (iw+1)] -------------------
    for (int e = threadIdx.x; e < NQ * LROW; e += 256) {
        const int row = e / LROW;                  // c*3 + kh
        const int col = e - row * LROW;            // iw + 1
        const int c   = row / 3;
        const int khr = row - 3 * c;
        const int ih  = oh - 1 + khr;
        const int iw  = col - 1;
        const bool inb = ((unsigned)ih < HW) & ((unsigned)iw < HW);
        lds1[e] = inb ? x[((b * C_IN + c) * HW + ih) * HW + iw] : 0.0f;
    }
    __syncthreads();
    // --- expand to k-major per column: lds2[ow*290 + k] = P[k][l] --------
    {
        const int owc = threadIdx.x >> 3;          // 0..31
        const int sub = threadIdx.x & 7;           // 0..7
#pragma unroll
        for (int i = 0; i < 12; ++i) {
            const int kq = sub + i * 8;            // 0..95
            const int s1 = kq * LROW + owc;
            const int s2 = owc * BSTR + kq * 3;
            lds2[s2 + 0] = lds1[s1 + 0];
            lds2[s2 + 1] = lds1[s1 + 1];
            lds2[s2 + 2] = lds1[s1 + 2];
        }
    }
    __syncthreads();
}

// 72 straight-line WMMAs; per j: batch 16 operand loads, then 8 WMMAs.
template <typename F>
__device__ __forceinline__ void wave_tiles(const float* abase,
                                           const float* bbase, F&& f) {
#pragma unroll
    for (int j = 0; j < NBLK; ++j) {
        v2f A[8], B[8];
#pragma unroll
        for (int kk = 0; kk < 8; ++kk) {
            const int t = j * 8 + kk;                      // compile-time
            A[kk] = *(const v2f*)(abase + 4 * t);          // ds_load_b64
            B[kk] = *(const v2f*)(bbase + 4 * t);          // ds_load_b64
        }
        v8f c = {0.f, 0.f, 0.f, 0.f, 0.f, 0.f, 0.f, 0.f};
#pragma unroll
        for (int kk = 0; kk < 8; ++kk)
            c = __builtin_amdgcn_wmma_f32_16x16x4_f32(
                    false, A[kk], false, B[kk], (short)0, c, false, false);
        f(j, c);
    }
}

__global__ void k_init_max(unsigned int* maxbits) { *maxbits = 0u; }

__global__ __launch_bounds__(256)
void k_pass1_max(const float* __restrict__ x, const float* __restrict__ w,
                 unsigned int* __restrict__ maxbits) {
    extern __shared__ float smem[];
    float* wlds = smem + WOFF;
    float* lds2 = smem + BOFF;
    float* lds1 = smem + HOFF;
    const int b  = blockIdx.x >> 5;
    const int oh = blockIdx.x & 31;
    stage_all(wlds, lds1, lds2, x, w, b, oh);

    const int wv   = threadIdx.x >> 5;
    const int lane = threadIdx.x & 31;
    const int half = lane >> 4;
    const int mn   = lane & 15;
    const int o0   = (wv >> 1) << 4;                 // M-tile
    const int ow   = ((wv & 1) << 4) + mn;           // N column
    const float* abase = wlds + (o0 + mn) * WSTR + 2 * half;
    const float* bbase = lds2 + ow * BSTR + 2 * half;

    float m = 0.0f;
    wave_tiles(abase, bbase, [&](int j, v8f c) {
        (void)j;
#pragma unroll
        for (int g = 0; g < 8; ++g) m = fmaxf(m, fabsf(c[g]));
    });

#pragma unroll
    for (int off = 16; off > 0; off >>= 1)
        m = fmaxf(m, __shfl_xor(m, off, 32));
    if (lane == 0)
        atomicMax(maxbits, __float_as_uint(m));      // valid: m >= 0
}

__global__ __launch_bounds__(256)
void k_pass2_out(const float* __restrict__ x, const float* __restrict__ w,
                 const unsigned int* __restrict__ maxbits,
                 float* __restrict__ out) {
    extern __shared__ float smem[];
    float* wlds = smem + WOFF;
    float* lds2 = smem + BOFF;
    float* lds1 = smem + HOFF;
    const int b  = blockIdx.x >> 5;
    const int oh = blockIdx.x & 31;
    stage_all(wlds, lds1, lds2, x, w, b, oh);

    const int wv   = threadIdx.x >> 5;
    const int lane = threadIdx.x & 31;
    const int half = lane >> 4;
    const int mn   = lane & 15;
    const int o0   = (wv >> 1) << 4;
    const int ow   = ((wv & 1) << 4) + mn;
    const float* abase = wlds + (o0 + mn) * WSTR + 2 * half;
    const float* bbase = lds2 + ow * BSTR + 2 * half;

    const float mx    = __uint_as_float(*maxbits);
    const float scale = fmaxf(mx * (1.0f / 127.0f), 1e-8f);

    float acc[8];
#pragma unroll
    for (int g = 0; g < 8; ++g) acc[g] = 0.0f;

    wave_tiles(abase, bbase, [&](int j, v8f c) {
        (void)j;
#pragma unroll
        for (int g = 0; g < 8; ++g)
            acc[g] += rintf(c[g] / scale) * scale;   // RNE == jnp.round
    });

    // D layout: VGPR g -> M = g + 8*half; l = oh*32 + ow
    const int lidx = oh * HW + ow;
#pragma unroll
    for (int g = 0; g < 8; ++g) {
        const int o = o0 + g + 8 * half;
        out[(b * OCH + o) * LDIM + lidx] = acc[g];
    }
}

extern "C" void kernel_launch(void* const* d_in, const int* in_sizes, int n_in,
                              void* d_out, int out_size, void* d_ws, size_t ws_size,
                              hipStream_t stream) {
    const float* x = (const float*)d_in[0];   // (4,32,32,32) f32
    const float* w = (const float*)d_in[1];   // (64,32,3,3)  f32
    float* out = (float*)d_out;               // (4,64,32,32) f32
    unsigned int* maxbits = (unsigned int*)d_ws;

    // 128 blocks = (b, oh); 8 waves/block = 4 M-tiles x 2 N-tiles.
    k_init_max<<<1, 1, 0, stream>>>(maxbits);
    k_pass1_max<<<128, 256, SMEM_BYTES, stream>>>(x, w, maxbits);
    k_pass2_out<<<128, 256, SMEM_BYTES, stream>>>(x, w, maxbits, out);
}